// SpectralCapsules_79267916415235
// MI455X (gfx1250) — compile-verified
//
#include <hip/hip_runtime.h>

typedef __attribute__((ext_vector_type(2))) float v2f;
typedef __attribute__((ext_vector_type(4))) float v4f;
typedef __attribute__((ext_vector_type(8))) float v8f;

#define BATCH 16
#define MPOS  256      // H*W
#define LCAP  32       // in capsules
#define OCAP  32       // out capsules
#define CCH   544      // channels = L*(P+1)
#define ACH   512      // activation channel base = L*P
#define PITER 48       // power iterations

__global__ __launch_bounds__(256) void speccaps_fused_kernel(
    const float* __restrict__ in,     // [B, M, 544]
    const float* __restrict__ wgt,    // [L, O, 4, 4]
    const float* __restrict__ bias,   // [O]
    float* __restrict__ out)          // [512 act | 8192 pose]
{
    __shared__ float sW[LCAP * 16];   // W[:, o] : sW[l*16 + j*4 + k]
    __shared__ float sG[256];         // 16x16 gram accumulator
    __shared__ float sU[16];          // eigenvector iterate

    const int tid  = threadIdx.x;
    const int blk  = blockIdx.x;
    const int b    = blk >> 5;
    const int o    = blk & 31;

    // Stage this output-capsule's transform weights into LDS.
    for (int s = tid; s < LCAP * 16; s += 256) {
        const int l = s >> 4;
        sW[s] = wgt[(l * OCAP + o) * 16 + (s & 15)];
    }
    sG[tid] = 0.0f;
    __syncthreads();

    const int wave = tid >> 5;
    const int lane = tid & 31;
    const int col  = lane & 15;   // gram column / P component
    const int half = lane >> 4;   // 0: rows {m0,m0+1}, 1: rows {m0+2,m0+3}
    const int i4   = col >> 2;    // pose row index i
    const int kk   = col & 3;     // weight column index k

    v8f acc0 = {0.f,0.f,0.f,0.f,0.f,0.f,0.f,0.f};
    v8f acc1 = {0.f,0.f,0.f,0.f,0.f,0.f,0.f,0.f};

    const float* inb = in + (size_t)b * MPOS * CCH;

    for (int ll = 0; ll < 4; ++ll) {
        const int l = wave * 4 + ll;
        // W column for this lane's (j=0..3, k=kk)
        const float w0 = sW[l * 16 + 0 * 4 + kk];
        const float w1 = sW[l * 16 + 1 * 4 + kk];
        const float w2 = sW[l * 16 + 2 * 4 + kk];
        const float w3 = sW[l * 16 + 3 * 4 + kk];
        const float* pl = inb + l * 16 + i4 * 4;  // pose row i of (m, l)
        const float* al = inb + ACH + l;          // activation of (m, l)

        for (int m0 = 0; m0 < MPOS; m0 += 8) {
            // ---- rows m0 .. m0+3 -> acc0 ----
            {
                const int mA = m0 + half * 2;
                const int mB = mA + 1;
                __builtin_prefetch(pl + (size_t)(mA + 8) * CCH, 0, 3);
                const v4f pA = *(const v4f*)(pl + (size_t)mA * CCH);
                const v4f pB = *(const v4f*)(pl + (size_t)mB * CCH);
                const float aA = al[(size_t)mA * CCH];
                const float aB = al[(size_t)mB * CCH];
                v2f y;
                y.x = aA * (pA.x * w0 + pA.y * w1 + pA.z * w2 + pA.w * w3);
                y.y = aB * (pB.x * w0 + pB.y * w1 + pB.z * w2 + pB.w * w3);
                // Symmetric rank-4 update: same registers serve as A (16x4) and B (4x16).
                acc0 = __builtin_amdgcn_wmma_f32_16x16x4_f32(
                           false, y, false, y, (short)0, acc0, false, false);
            }
            // ---- rows m0+4 .. m0+7 -> acc1 ----
            {
                const int mA = m0 + 4 + half * 2;
                const int mB = mA + 1;
                __builtin_prefetch(pl + (size_t)(mA + 8) * CCH, 0, 3);
                const v4f pA = *(const v4f*)(pl + (size_t)mA * CCH);
                const v4f pB = *(const v4f*)(pl + (size_t)mB * CCH);
                const float aA = al[(size_t)mA * CCH];
                const float aB = al[(size_t)mB * CCH];
                v2f y;
                y.x = aA * (pA.x * w0 + pA.y * w1 + pA.z * w2 + pA.w * w3);
                y.y = aB * (pB.x * w0 + pB.y * w1 + pB.z * w2 + pB.w * w3);
                acc1 = __builtin_amdgcn_wmma_f32_16x16x4_f32(
                           false, y, false, y, (short)0, acc1, false, false);
            }
        }
    }

    // Merge wave-partial grams into LDS (D layout: vgpr r -> row r + 8*half).
#pragma unroll
    for (int r = 0; r < 8; ++r) {
        const int row = r + half * 8;
        atomicAdd(&sG[row * 16 + col], acc0[r] + acc1[r]);
    }
    __syncthreads();

    if (wave == 0) {
        const int p = lane & 15;   // both lane halves duplicate the work

        if (lane < 16) sU[lane] = 0.25f;

        // Power iteration for the top eigenvector of the 16x16 gram.
        for (int it = 0; it < PITER; ++it) {
            float v = 0.0f;
#pragma unroll
            for (int q = 0; q < 16; ++q) v += sG[p * 16 + q] * sU[q];
            float s = v * v;
            s += __shfl_xor(s, 1);
            s += __shfl_xor(s, 2);
            s += __shfl_xor(s, 4);
            s += __shfl_xor(s, 8);
            v *= rsqrtf(s + 1e-30f);
            if (lane < 16) sU[lane] = v;
        }

        // ratio = (u^T G u) / trace(G)   (no second pass over Y needed)
        float gu = 0.0f;
#pragma unroll
        for (int q = 0; q < 16; ++q) gu += sG[p * 16 + q] * sU[q];
        float lam = gu * sU[p];
        float tr  = sG[p * 17];
        lam += __shfl_xor(lam, 1);
        lam += __shfl_xor(lam, 2);
        lam += __shfl_xor(lam, 4);
        lam += __shfl_xor(lam, 8);
        tr  += __shfl_xor(tr, 1);
        tr  += __shfl_xor(tr, 2);
        tr  += __shfl_xor(tr, 4);
        tr  += __shfl_xor(tr, 8);

        const float ratio = lam / tr;
        const float act = 1.0f / (1.0f + __expf(-(ratio - bias[o])));

        if (lane == 0) out[b * OCAP + o] = act;
        if (lane < 16) out[BATCH * OCAP + (b * OCAP + o) * 16 + lane] = sU[lane];
    }
}

extern "C" void kernel_launch(void* const* d_in, const int* in_sizes, int n_in,
                              void* d_out, int out_size, void* d_ws, size_t ws_size,
                              hipStream_t stream) {
    const float* in   = (const float*)d_in[0];
    const float* wgt  = (const float*)d_in[1];
    const float* bias = (const float*)d_in[2];
    float* out = (float*)d_out;

    dim3 grid(BATCH * OCAP);   // 512 blocks, one per (b, o)
    dim3 block(256);           // 8 wave32
    hipLaunchKernelGGL(speccaps_fused_kernel, grid, block, 0, stream,
                       in, wgt, bias, out);
}